// DMoN_9801115369798
// MI455X (gfx1250) — compile-verified
//
#include <hip/hip_runtime.h>
#include <math.h>

typedef __attribute__((ext_vector_type(2))) float v2f;
typedef __attribute__((ext_vector_type(8))) float v8f;

#define DDIM 256
#define KC   16

// Single-instruction lane-xor shuffle via ds_swizzle (group-of-32 encoding:
// offset[14:10]=xor_mask, offset[9:5]=or_mask=0, offset[4:0]=and_mask=0x1f).
// Masks 1/2/4/8 stay within a 16-lane half; mask 16 combines the two halves.
#define SWZ(x, m) __int_as_float(__builtin_amdgcn_ds_swizzle(__float_as_int(x), (((m) << 10) | 0x1f)))

// ---------------- workspace layout (floats) ----------------
// [0      .. 4095] : P = S^T X accumulator (16 x 256)
// [4096   .. 4111] : cluster_sizes[16]
// [4112   .. 4127] : left[16]  (S^T d)
// [4128]           : m2
// [4129]           : traceAS
#define WS_P      0
#define WS_SIZES  4096
#define WS_ACC    4112   // left[16], then m2 at +16, trace at +17
#define WS_COUNT  4130

__global__ void k_zero(float* __restrict__ ws, int cnt) {
    int i = blockIdx.x * blockDim.x + threadIdx.x;
    if (i < cnt) ws[i] = 0.0f;
}

// ---- Kernel: logits = X @ W^T + b ; softmax over K=16 ; write S ----
// One wave handles a 16-row tile of X.  WMMA f32 16x16x4, 64 steps over D=256.
__global__ void k_assign(const float* __restrict__ X, const float* __restrict__ W,
                         const float* __restrict__ bias, float* __restrict__ S, int n) {
    const int lane = threadIdx.x & 31;
    const int hh   = lane >> 4;     // half-wave id
    const int lm   = lane & 15;
    const int tile = blockIdx.x * (blockDim.x >> 5) + (threadIdx.x >> 5);
    const int row0 = tile * 16;
    if (row0 >= n) return;

    int arow = row0 + lm;
    if (arow >= n) arow = n - 1;                    // clamp (stores are masked below)
    const float* ap = X + (size_t)arow * DDIM + 2 * hh;   // A: X[row][d+2h .. +1]
    const float* bp = W + (size_t)lm   * DDIM + 2 * hh;   // B: W[cluster][d+2h .. +1]

    v8f c = {};
    #pragma unroll 4
    for (int d = 0; d < DDIM; d += 4) {
        v2f a  = *(const v2f*)(ap + d);
        v2f bb = *(const v2f*)(bp + d);
        c = __builtin_amdgcn_wmma_f32_16x16x4_f32(false, a, false, bb,
                                                  (short)0, c, false, false);
    }

    const float bv = bias[lm];
    #pragma unroll
    for (int r = 0; r < 8; ++r) {
        float x = c[r] + bv;                         // logits[row0+r+8h][lm]
        float m = x;
        m = fmaxf(m, SWZ(m, 1));
        m = fmaxf(m, SWZ(m, 2));
        m = fmaxf(m, SWZ(m, 4));
        m = fmaxf(m, SWZ(m, 8));
        float e = __expf(x - m);
        float s = e;
        s += SWZ(s, 1);
        s += SWZ(s, 2);
        s += SWZ(s, 4);
        s += SWZ(s, 8);
        float sv = e / s;
        int nr = row0 + r + 8 * hh;
        if (nr < n) S[(size_t)nr * KC + lm] = sv;
    }
}

// ---- Kernel: P = S^T @ X  (16 x 256, reduce over N) + cluster_sizes ----
// grid.x = 16 D-tiles, grid.y = blocks of 8 waves; each wave strides node-groups of 4.
__global__ void k_pool(const float* __restrict__ X, const float* __restrict__ S,
                       float* __restrict__ P, float* __restrict__ sizes, int n) {
    const int lane = threadIdx.x & 31;
    const int hh   = lane >> 4;
    const int lm   = lane & 15;
    const int d0   = blockIdx.x * 16;
    const int wv   = blockIdx.y * (blockDim.x >> 5) + (threadIdx.x >> 5);
    const int wtot = gridDim.y * (blockDim.x >> 5);
    const int G    = (n + 3) >> 2;

    v8f c = {};
    float sz = 0.0f;
    for (int g = wv; g < G; g += wtot) {
        const int r0 = g * 4 + 2 * hh;
        const int r1 = r0 + 1;
        float a0 = 0.f, a1 = 0.f, b0 = 0.f, b1 = 0.f;
        if (r0 < n) { a0 = S[(size_t)r0 * KC + lm]; b0 = X[(size_t)r0 * DDIM + d0 + lm]; }
        if (r1 < n) { a1 = S[(size_t)r1 * KC + lm]; b1 = X[(size_t)r1 * DDIM + d0 + lm]; }
        v2f a  = {a0, a1};   // A = S^T : M=cluster(lm), K=node
        v2f bb = {b0, b1};   // B = X   : K=node, N=dcol(lm)
        c = __builtin_amdgcn_wmma_f32_16x16x4_f32(false, a, false, bb,
                                                  (short)0, c, false, false);
        sz += a0 + a1;       // cluster_sizes partial: column lm
    }

    if (blockIdx.x == 0) {   // only one D-tile contributes sizes (S read once there)
        sz += SWZ(sz, 16);
        if (lane < 16) atomicAdd(&sizes[lm], sz);
    }
    #pragma unroll
    for (int r = 0; r < 8; ++r)
        atomicAdd(&P[(size_t)(r + 8 * hh) * DDIM + d0 + lm], c[r]);
}

// ---- Kernel: edge pass -> trace(S^T A S), left = S^T d, m2 ----
// Half-wave per edge: lane lm handles cluster component lm (64B coalesced gathers from L2).
__global__ void k_edge(const int* __restrict__ ei, const float* __restrict__ adj,
                       const float* __restrict__ S, float* __restrict__ acc, int E_) {
    __shared__ float red[18];
    const int t = threadIdx.x;
    if (t < 18) red[t] = 0.0f;
    __syncthreads();

    const int lane = t & 31;
    const int hh   = lane >> 4;
    const int lm   = lane & 15;
    const int wv   = blockIdx.x * (blockDim.x >> 5) + (t >> 5);
    const int wtot = gridDim.x * (blockDim.x >> 5);
    const int* rows = ei;
    const int* cols = ei + E_;

    float lacc = 0.f, tacc = 0.f, macc = 0.f;
    for (int e = wv * 2 + hh; e < E_; e += wtot * 2) {
        const int r = rows[e];
        const int q = cols[e];
        const float a = adj[e];
        const float sr = S[(size_t)r * KC + lm];
        const float sc = S[(size_t)q * KC + lm];
        float p = sr * sc;
        p += SWZ(p, 1);
        p += SWZ(p, 2);
        p += SWZ(p, 4);
        p += SWZ(p, 8);                     // p = dot(S[r], S[c]) on all 16 lanes
        lacc += a * sr;                     // left[lm] partial
        if (lm == 0) { tacc += a * p; macc += a; }
    }
    lacc += SWZ(lacc, 16);
    if (lane < 16) atomicAdd(&red[lm], lacc);
    tacc += SWZ(tacc, 16);
    macc += SWZ(macc, 16);
    if (lane == 0) { atomicAdd(&red[16], macc); atomicAdd(&red[17], tacc); }
    __syncthreads();
    if (t < 18) atomicAdd(&acc[t], red[t]);
}

// ---- Kernel: finalize -> selu(P / sizes), spectral_loss, collapse_loss ----
__global__ void k_fin(const float* __restrict__ ws, float* __restrict__ out, int n) {
    const float* P     = ws + WS_P;
    const float* sizes = ws + WS_SIZES;
    const float* left  = ws + WS_ACC;
    const float  m2    = ws[WS_ACC + 16];
    const float  tr    = ws[WS_ACC + 17];

    float nrm2 = 0.f, sl = 0.f;
    #pragma unroll
    for (int k = 0; k < KC; ++k) { nrm2 += sizes[k] * sizes[k]; sl += left[k] * left[k]; }
    const float spectral = -(tr - sl / m2) / m2;
    const float collapse = 0.1f * (sqrtf(nrm2) / (float)n * sqrtf((float)KC) - 1.0f);

    const float scale = 1.0507009873554805f;
    const float alpha = 1.6732632423543772f;
    for (int i = threadIdx.x; i < KC * DDIM; i += blockDim.x) {
        const int k = i >> 8;                // i / DDIM
        const float x = P[i] / sizes[k];
        out[i] = (x > 0.0f) ? scale * x : scale * alpha * (expf(x) - 1.0f);
    }
    if (threadIdx.x == 0) {
        size_t off = (size_t)KC * DDIM + (size_t)n * KC;
        out[off]     = spectral;
        out[off + 1] = collapse;
    }
}

extern "C" void kernel_launch(void* const* d_in, const int* in_sizes, int n_in,
                              void* d_out, int out_size, void* d_ws, size_t ws_size,
                              hipStream_t stream) {
    const float* X    = (const float*)d_in[0];
    const int*   ei   = (const int*)d_in[1];
    const float* adj  = (const float*)d_in[2];
    const float* W    = (const float*)d_in[3];
    const float* bias = (const float*)d_in[4];
    float* out = (float*)d_out;
    float* ws  = (float*)d_ws;

    const int n  = in_sizes[0] / DDIM;   // N = 100000
    const int E_ = in_sizes[2];          // E = 3200000
    float* S = out + KC * DDIM;          // assignments live directly in d_out

    k_zero<<<(WS_COUNT + 255) / 256, 256, 0, stream>>>(ws, WS_COUNT);

    const int tiles = (n + 15) / 16;
    k_assign<<<(tiles + 3) / 4, 128, 0, stream>>>(X, W, bias, S, n);

    dim3 g3(KC, 32);                     // 16 D-tiles x 32 blocks of 8 waves
    k_pool<<<g3, 256, 0, stream>>>(X, S, ws + WS_P, ws + WS_SIZES, n);

    k_edge<<<1024, 256, 0, stream>>>(ei, adj, S, ws + WS_ACC, E_);

    k_fin<<<1, 256, 0, stream>>>(ws, out, n);
}